// GNN_79465484910640
// MI455X (gfx1250) — compile-verified
//
#include <hip/hip_runtime.h>
#include <math.h>

typedef __attribute__((ext_vector_type(2))) float v2f;
typedef __attribute__((ext_vector_type(8))) float v8f;

// ---------------------------------------------------------------------------
// Fused triple-GEMM + epilogue using V_WMMA_F32_16X16X4_F32 (fp32-exact).
//   Aout[m,n]  = (h@W1)[m,n] + b1[n]
//   Outi[m,n]  = (h@W3)[m,n] + b3[n] - (h@W2)[m,n] * deg[m]
// DIN in {128, 64}, DOUT fixed 64. One wave per 16-row tile.
// ---------------------------------------------------------------------------
template <int DIN>
__global__ __launch_bounds__(256) void gemm3_wmma(
    const float* __restrict__ h,
    const float* __restrict__ W1, const float* __restrict__ b1,
    const float* __restrict__ W2,
    const float* __restrict__ W3, const float* __restrict__ b3,
    const float* __restrict__ deg,
    float* __restrict__ Aout, float* __restrict__ Outi, int n)
{
    constexpr int KS = DIN / 4;              // K-steps of 4
    const int wid  = threadIdx.x >> 5;       // wave id in block (8 waves)
    const int rt   = blockIdx.x * 8 + wid;   // 16-row tile index
    const int m0   = rt * 16;
    if (m0 >= n) return;                     // wave-uniform -> EXEC stays all-1

    const int lane = threadIdx.x & 31;
    const int r    = lane & 15;              // row within tile / col within tile
    const int half = lane >> 4;              // 0 or 1
    const int kb   = half * 2;               // K pair base (0 or 2)

    int gr = m0 + r; if (gr > n - 1) gr = n - 1;

    // A fragments for the whole K range: lane holds (row=r, K=ks*4+kb, +1)
    v2f afrag[KS];
    const float* hrow = h + (size_t)gr * DIN;
#pragma unroll
    for (int i = 0; i < KS; ++i)
        afrag[i] = *(const v2f*)(hrow + i * 4 + kb);

#pragma unroll 1
    for (int nt = 0; nt < 4; ++nt) {
        const int col = nt * 16 + r;
        v8f acc1 = {}; v8f acc2 = {}; v8f acc3 = {};
#pragma unroll
        for (int ks = 0; ks < KS; ++ks) {
            const int kk = ks * 4 + kb;
            v2f bf1, bf2, bf3;
            bf1.x = W1[(size_t)kk * 64 + col]; bf1.y = W1[(size_t)(kk + 1) * 64 + col];
            bf2.x = W2[(size_t)kk * 64 + col]; bf2.y = W2[(size_t)(kk + 1) * 64 + col];
            bf3.x = W3[(size_t)kk * 64 + col]; bf3.y = W3[(size_t)(kk + 1) * 64 + col];
            acc1 = __builtin_amdgcn_wmma_f32_16x16x4_f32(false, afrag[ks], false, bf1,
                                                         (short)0, acc1, false, false);
            acc2 = __builtin_amdgcn_wmma_f32_16x16x4_f32(false, afrag[ks], false, bf2,
                                                         (short)0, acc2, false, false);
            acc3 = __builtin_amdgcn_wmma_f32_16x16x4_f32(false, afrag[ks], false, bf3,
                                                         (short)0, acc3, false, false);
        }
        const float bias1 = b1[col];
        const float bias3 = b3[col];
#pragma unroll
        for (int v = 0; v < 8; ++v) {
            const int m = m0 + v + half * 8;    // C/D layout: VGPR v -> M=v (+8 hi half)
            if (m < n) {
                const float dg = deg[m];
                Aout[(size_t)m * 64 + col] = acc1[v] + bias1;
                Outi[(size_t)m * 64 + col] = acc3[v] + bias3 - acc2[v] * dg;
            }
        }
    }
}

// ---------------------------------------------------------------------------
// Edge phase: Out[dst] += ew * A[src]  (dout=64). 16 threads/edge, float4 each.
// A (25.6 MB) lives in L2 -> gathers stay on-chip.
// ---------------------------------------------------------------------------
__global__ __launch_bounds__(256) void scatter64(
    const int* __restrict__ src, const int* __restrict__ dst,
    const float* __restrict__ ew, const float* __restrict__ A,
    float* __restrict__ Out, int e)
{
    const long long tid = (long long)blockIdx.x * blockDim.x + threadIdx.x;
    const int eidx = (int)(tid >> 4);
    if (eidx >= e) return;
    const int c = (int)(tid & 15) * 4;
    const int s = src[eidx];
    const int d = dst[eidx];
    const float w = ew[eidx];
    const float4 av = *(const float4*)(A + (size_t)s * 64 + c);
    float* op = Out + (size_t)d * 64 + c;
    atomicAdd(op + 0, w * av.x);
    atomicAdd(op + 1, w * av.y);
    atomicAdd(op + 2, w * av.z);
    atomicAdd(op + 3, w * av.w);
}

__global__ __launch_bounds__(256) void fill0_k(float* __restrict__ p, int n)
{
    const int i = blockIdx.x * blockDim.x + threadIdx.x;
    if (i < n) p[i] = 0.0f;
}

__global__ __launch_bounds__(256) void deg_k(
    const int* __restrict__ dst, const float* __restrict__ ew,
    float* __restrict__ deg, int e)
{
    const int i = blockIdx.x * blockDim.x + threadIdx.x;
    if (i < e) atomicAdd(deg + dst[i], ew[i]);
}

__global__ __launch_bounds__(256) void leaky_ip(float* __restrict__ p, int n)
{
    const int i = blockIdx.x * blockDim.x + threadIdx.x;
    if (i < n) {
        const float v = p[i];
        p[i] = v > 0.0f ? v : 0.1f * v;
    }
}

// ---------------------------------------------------------------------------
// Layer 4: dout = 1. Plain dot products per node (WMMA pointless at N-dim 1).
// ---------------------------------------------------------------------------
__global__ __launch_bounds__(256) void lin_d1(
    const float* __restrict__ h,
    const float* __restrict__ W1, const float* __restrict__ b1,
    const float* __restrict__ W2,
    const float* __restrict__ W3, const float* __restrict__ b3,
    const float* __restrict__ deg,
    float* __restrict__ A4, float* __restrict__ Out4, int n)
{
    const int i = blockIdx.x * blockDim.x + threadIdx.x;
    if (i >= n) return;
    const float4* hv  = (const float4*)(h + (size_t)i * 64);
    const float4* w1v = (const float4*)W1;
    const float4* w2v = (const float4*)W2;
    const float4* w3v = (const float4*)W3;
    float s1 = 0.f, s2 = 0.f, s3 = 0.f;
#pragma unroll
    for (int j = 0; j < 16; ++j) {
        const float4 hh = hv[j];
        const float4 a = w1v[j], b = w2v[j], c = w3v[j];
        s1 += hh.x * a.x + hh.y * a.y + hh.z * a.z + hh.w * a.w;
        s2 += hh.x * b.x + hh.y * b.y + hh.z * b.z + hh.w * b.w;
        s3 += hh.x * c.x + hh.y * c.y + hh.z * c.z + hh.w * c.w;
    }
    A4[i]   = s1 + b1[0];
    Out4[i] = s3 + b3[0] - s2 * deg[i];
}

__global__ __launch_bounds__(256) void scatter1(
    const int* __restrict__ src, const int* __restrict__ dst,
    const float* __restrict__ ew, const float* __restrict__ A4,
    float* __restrict__ Out4, int e)
{
    const int i = blockIdx.x * blockDim.x + threadIdx.x;
    if (i < e) atomicAdd(Out4 + dst[i], ew[i] * A4[src[i]]);
}

__global__ __launch_bounds__(256) void final_act(
    const float* __restrict__ Out4, float* __restrict__ y, int n)
{
    const int i = blockIdx.x * blockDim.x + threadIdx.x;
    if (i < n) {
        float v = Out4[i];
        v = v > 0.0f ? v : 0.1f * v;
        y[i] = 1.0f / (1.0f + __expf(-v));
    }
}

// ---------------------------------------------------------------------------
extern "C" void kernel_launch(void* const* d_in, const int* in_sizes, int n_in,
                              void* d_out, int out_size, void* d_ws, size_t ws_size,
                              hipStream_t stream)
{
    const float* x  = (const float*)d_in[0];
    const int*   ei = (const int*)d_in[1];
    const float* ew = (const float*)d_in[2];
    const int E = in_sizes[2];
    const int N = in_sizes[0] / 128;
    const int* srcI = ei;
    const int* dstI = ei + E;

    // Param flattening order: dict insertion (W1,b1,W2,W3,b3) vs pytree-sorted
    // (W1,W2,W3,b1,b3). Disambiguate via size of slot 4 (64 => bias follows W1).
    const bool insertion = (in_sizes[4] < in_sizes[3]);
    const float* P[4][5];  // [layer][W1,b1,W2,W3,b3]
    for (int l = 0; l < 4; ++l) {
        const int base = 3 + l * 5;
        if (insertion) {
            P[l][0] = (const float*)d_in[base + 0];
            P[l][1] = (const float*)d_in[base + 1];
            P[l][2] = (const float*)d_in[base + 2];
            P[l][3] = (const float*)d_in[base + 3];
            P[l][4] = (const float*)d_in[base + 4];
        } else {
            P[l][0] = (const float*)d_in[base + 0];
            P[l][2] = (const float*)d_in[base + 1];
            P[l][3] = (const float*)d_in[base + 2];
            P[l][1] = (const float*)d_in[base + 3];
            P[l][4] = (const float*)d_in[base + 4];
        }
    }

    // Workspace: deg[N] | A[N*64] | O0[N*64] | O1[N*64]  (~77 MB)
    float* deg = (float*)d_ws;
    float* A   = deg + N;
    float* O0  = A  + (size_t)N * 64;
    float* O1  = O0 + (size_t)N * 64;

    const int gemmBlocks = ((N + 15) / 16 + 7) / 8;
    const int sBlocks    = (int)((16LL * E + 255) / 256);
    const int eBlocks    = (E + 255) / 256;
    const int nBlocks    = (N + 255) / 256;
    const int actBlocks  = (N * 64 + 255) / 256;

    // deg_w = segment_sum(ew, dst)
    fill0_k<<<nBlocks, 256, 0, stream>>>(deg, N);
    deg_k<<<eBlocks, 256, 0, stream>>>(dstI, ew, deg, E);

    // Layer 1: 128 -> 64
    gemm3_wmma<128><<<gemmBlocks, 256, 0, stream>>>(
        x, P[0][0], P[0][1], P[0][2], P[0][3], P[0][4], deg, A, O0, N);
    scatter64<<<sBlocks, 256, 0, stream>>>(srcI, dstI, ew, A, O0, E);
    leaky_ip<<<actBlocks, 256, 0, stream>>>(O0, N * 64);

    // Layer 2: 64 -> 64
    gemm3_wmma<64><<<gemmBlocks, 256, 0, stream>>>(
        O0, P[1][0], P[1][1], P[1][2], P[1][3], P[1][4], deg, A, O1, N);
    scatter64<<<sBlocks, 256, 0, stream>>>(srcI, dstI, ew, A, O1, E);
    leaky_ip<<<actBlocks, 256, 0, stream>>>(O1, N * 64);

    // Layer 3: 64 -> 64
    gemm3_wmma<64><<<gemmBlocks, 256, 0, stream>>>(
        O1, P[2][0], P[2][1], P[2][2], P[2][3], P[2][4], deg, A, O0, N);
    scatter64<<<sBlocks, 256, 0, stream>>>(srcI, dstI, ew, A, O0, E);
    leaky_ip<<<actBlocks, 256, 0, stream>>>(O0, N * 64);

    // Layer 4: 64 -> 1 (A buffer reused for a[N], O1 reused for out[N])
    lin_d1<<<nBlocks, 256, 0, stream>>>(
        O0, P[3][0], P[3][1], P[3][2], P[3][3], P[3][4], deg, A, O1, N);
    scatter1<<<eBlocks, 256, 0, stream>>>(srcI, dstI, ew, A, O1, E);
    final_act<<<nBlocks, 256, 0, stream>>>(O1, (float*)d_out, N);
}